// GraphClassifier_79293686219286
// MI455X (gfx1250) — compile-verified
//
#include <hip/hip_runtime.h>
#include <hip/hip_bf16.h>
#include <stdint.h>

typedef float f32x2 __attribute__((ext_vector_type(2)));
typedef float f32x8 __attribute__((ext_vector_type(8)));

#define DDIM 128
#define EPSBN 1e-5f

// ---------------------------------------------------------------------------
// small utility kernels
// ---------------------------------------------------------------------------
__global__ void fill_f32_kernel(float* p, float v, int n) {
    int i = blockIdx.x * blockDim.x + threadIdx.x;
    if (i < n) p[i] = v;
}

__global__ void deg_accum_kernel(const int* __restrict__ dst, float* __restrict__ deg, int e) {
    int i = blockIdx.x * blockDim.x + threadIdx.x;
    if (i < e) atomicAdd(&deg[dst[i]], 1.0f);
}

__global__ void deg_rsqrt_kernel(float* __restrict__ deg, int n) {
    int i = blockIdx.x * blockDim.x + threadIdx.x;
    if (i < n) deg[i] = rsqrtf(fmaxf(deg[i], 1.0f));   // dinv, in place
}

// ---------------------------------------------------------------------------
// GEMM: Out[N x 128] = A[N x 128] @ W[128 x 128], fp32 WMMA 16x16x4.
// One wave -> one 16-row stripe across all 128 output columns.
// W staged in LDS via GLOBAL_LOAD_ASYNC_TO_LDS_B128 (no VGPR round-trip),
// drained with s_wait_asynccnt before the workgroup barrier.
// ---------------------------------------------------------------------------
__global__ __launch_bounds__(256) void gemm128_wmma_kernel(
    const float* __restrict__ A, const float* __restrict__ W,
    float* __restrict__ Out, int nrows) {
    __shared__ float Wlds[DDIM * DDIM];
    const int t = threadIdx.x;

    // --- async stage of W (64 KB): 256 threads x 16 x b128 copies ---
    {
        unsigned lds_base = (unsigned)(uintptr_t)(&Wlds[0]);  // LDS aperture low bits
        const char* wbytes = (const char*)W;
#pragma unroll
        for (int r = 0; r < 16; ++r) {
            unsigned byte_off = (unsigned)(r * 4096 + t * 16);
            unsigned lds_addr = lds_base + byte_off;
            const void* gaddr = (const void*)(wbytes + byte_off);
            asm volatile("global_load_async_to_lds_b128 %0, %1, off"
                         :: "v"(lds_addr), "v"(gaddr)
                         : "memory");
        }
        asm volatile("s_wait_asynccnt 0" ::: "memory");
    }
    __syncthreads();

    const int wave = t >> 5;                 // 0..7
    const int lane = t & 31;
    const int m0 = (blockIdx.x * 8 + wave) * 16;
    if (m0 >= nrows) return;                 // wave-uniform: EXEC all-1 inside

    const int half = lane >> 4;              // 0: K base 0, 1: K base 2
    const int l16  = lane & 15;
    const int kb   = half * 2;

    f32x8 acc[8] = {};
    const float* arow = A + (size_t)(m0 + l16) * DDIM;

    for (int k = 0; k < DDIM; k += 4) {
        f32x2 a;
        a.x = arow[k + kb];
        a.y = arow[k + kb + 1];
#pragma unroll
        for (int n = 0; n < 8; ++n) {
            f32x2 b;
            b.x = Wlds[(k + kb)     * DDIM + n * 16 + l16];
            b.y = Wlds[(k + kb + 1) * DDIM + n * 16 + l16];
            acc[n] = __builtin_amdgcn_wmma_f32_16x16x4_f32(
                false, a, false, b, (short)0, acc[n], false, false);
        }
    }
    // D layout: VGPR j -> row m0 + j + half*8, col n*16 + l16
#pragma unroll
    for (int n = 0; n < 8; ++n) {
#pragma unroll
        for (int j = 0; j < 8; ++j) {
            Out[(size_t)(m0 + j + half * 8) * DDIM + n * 16 + l16] = acc[n][j];
        }
    }
}

// ---------------------------------------------------------------------------
// out[i,:] = h[i,:]*dinv[i]^2 + bias   (self-loop contribution + bias)
// ---------------------------------------------------------------------------
__global__ void selfloop_init_kernel(const float* __restrict__ h,
                                     const float* __restrict__ dinv,
                                     const float* __restrict__ bias,
                                     float* __restrict__ out, int n) {
    int i = blockIdx.x * blockDim.x + threadIdx.x;   // over n*32 float4s
    if (i >= n * (DDIM / 4)) return;
    int row  = i >> 5;
    int col4 = i & 31;
    float di = dinv[row];
    float s  = di * di;
    float4 v = *(const float4*)(&h[(size_t)row * DDIM + col4 * 4]);
    float4 bb = *(const float4*)(&bias[col4 * 4]);
    float4 r;
    r.x = v.x * s + bb.x; r.y = v.y * s + bb.y;
    r.z = v.z * s + bb.z; r.w = v.w * s + bb.w;
    *(float4*)(&out[(size_t)row * DDIM + col4 * 4]) = r;
}

// ---------------------------------------------------------------------------
// edge scatter: one wave per edge, lane handles float4 (32 lanes * 4 = 128)
// out[dst,:] += h[src,:] * dinv[src]*dinv[dst]   via global_atomic_add_f32
// ---------------------------------------------------------------------------
__global__ __launch_bounds__(256) void scatter_kernel(
    const float* __restrict__ h, const int* __restrict__ src,
    const int* __restrict__ dst, const float* __restrict__ dinv,
    float* __restrict__ out, int e) {
    int wid  = blockIdx.x * (blockDim.x >> 5) + (threadIdx.x >> 5);
    int lane = threadIdx.x & 31;
    if (wid >= e) return;
    int s = src[wid], d = dst[wid];
    float nrm = dinv[s] * dinv[d];
    float4 v = *(const float4*)(&h[(size_t)s * DDIM + lane * 4]);
    float* o = &out[(size_t)d * DDIM + lane * 4];
    atomicAdd(o + 0, v.x * nrm);
    atomicAdd(o + 1, v.y * nrm);
    atomicAdd(o + 2, v.z * nrm);
    atomicAdd(o + 3, v.w * nrm);
}

__global__ void relu_kernel(float* __restrict__ x, int n4) {
    int i = blockIdx.x * blockDim.x + threadIdx.x;
    if (i >= n4) return;
    float4 v = *(float4*)(&x[i * 4]);
    v.x = fmaxf(v.x, 0.f); v.y = fmaxf(v.y, 0.f);
    v.z = fmaxf(v.z, 0.f); v.w = fmaxf(v.w, 0.f);
    *(float4*)(&x[i * 4]) = v;
}

// ---------------------------------------------------------------------------
// BatchNorm: column sums & sumsq (blockDim=128, one col per thread, coalesced)
// ---------------------------------------------------------------------------
#define BN_ROWS 256
__global__ __launch_bounds__(128) void bn_stats_kernel(
    const float* __restrict__ x, float* __restrict__ stats, int n) {
    int col = threadIdx.x;
    int r0  = blockIdx.x * BN_ROWS;
    int r1  = min(r0 + BN_ROWS, n);
    float s = 0.f, ss = 0.f;
    for (int r = r0; r < r1; ++r) {
        float v = x[(size_t)r * DDIM + col];
        s += v; ss += v * v;
    }
    atomicAdd(&stats[col], s);
    atomicAdd(&stats[DDIM + col], ss);
}

__global__ void bn_coef_kernel(const float* __restrict__ stats,
                               const float* __restrict__ g,
                               const float* __restrict__ be,
                               float* __restrict__ coef, int n) {
    int col = threadIdx.x;
    if (col >= DDIM) return;
    float inv = 1.0f / (float)n;
    float mu  = stats[col] * inv;
    float var = stats[DDIM + col] * inv - mu * mu;
    float sc  = g[col] * rsqrtf(var + EPSBN);
    coef[col]        = sc;
    coef[DDIM + col] = be[col] - mu * sc;
}

__global__ void bn_apply_relu_kernel(float* __restrict__ x,
                                     const float* __restrict__ coef, int n) {
    int i = blockIdx.x * blockDim.x + threadIdx.x;   // over n*32 float4s
    if (i >= n * (DDIM / 4)) return;
    int row  = i >> 5;
    int col4 = i & 31;
    float4 v  = *(float4*)(&x[(size_t)row * DDIM + col4 * 4]);
    float4 sc = *(const float4*)(&coef[col4 * 4]);
    float4 sh = *(const float4*)(&coef[DDIM + col4 * 4]);
    v.x = fmaxf(v.x * sc.x + sh.x, 0.f);
    v.y = fmaxf(v.y * sc.y + sh.y, 0.f);
    v.z = fmaxf(v.z * sc.z + sh.z, 0.f);
    v.w = fmaxf(v.w * sc.w + sh.w, 0.f);
    *(float4*)(&x[(size_t)row * DDIM + col4 * 4]) = v;
}

// ---------------------------------------------------------------------------
// mean pooling: one wave per node
// ---------------------------------------------------------------------------
__global__ __launch_bounds__(256) void pool_kernel(
    const float* __restrict__ h, const int* __restrict__ batch,
    float* __restrict__ pooled, float* __restrict__ cnt, int n) {
    int node = blockIdx.x * (blockDim.x >> 5) + (threadIdx.x >> 5);
    int lane = threadIdx.x & 31;
    if (node >= n) return;
    int g = batch[node];
    float4 v = *(const float4*)(&h[(size_t)node * DDIM + lane * 4]);
    float* o = &pooled[(size_t)g * DDIM + lane * 4];
    atomicAdd(o + 0, v.x);
    atomicAdd(o + 1, v.y);
    atomicAdd(o + 2, v.z);
    atomicAdd(o + 3, v.w);
    if (lane == 0) atomicAdd(&cnt[g], 1.0f);
}

// ---------------------------------------------------------------------------
// head: logits = (pooled/cnt) @ fc_w + fc_b ; log_softmax. one thread/graph
// ---------------------------------------------------------------------------
#define NCLS 10
__global__ void head_kernel(const float* __restrict__ pooled,
                            const float* __restrict__ cnt,
                            const float* __restrict__ fcw,
                            const float* __restrict__ fcb,
                            float* __restrict__ out, int gtot) {
    int g = blockIdx.x * blockDim.x + threadIdx.x;
    if (g >= gtot) return;
    float inv = 1.0f / fmaxf(cnt[g], 1.0f);
    float logits[NCLS];
#pragma unroll
    for (int c = 0; c < NCLS; ++c) logits[c] = fcb[c];
    for (int k = 0; k < DDIM; ++k) {
        float p = pooled[(size_t)g * DDIM + k] * inv;
#pragma unroll
        for (int c = 0; c < NCLS; ++c) logits[c] += p * fcw[k * NCLS + c];
    }
    float m = logits[0];
#pragma unroll
    for (int c = 1; c < NCLS; ++c) m = fmaxf(m, logits[c]);
    float s = 0.f;
#pragma unroll
    for (int c = 0; c < NCLS; ++c) s += __expf(logits[c] - m);
    float lse = __logf(s) + m;
#pragma unroll
    for (int c = 0; c < NCLS; ++c) out[g * NCLS + c] = logits[c] - lse;
}

// ---------------------------------------------------------------------------
extern "C" void kernel_launch(void* const* d_in, const int* in_sizes, int n_in,
                              void* d_out, int out_size, void* d_ws, size_t ws_size,
                              hipStream_t stream) {
    const float* x     = (const float*)d_in[0];
    const int*   ei    = (const int*)d_in[1];
    const int*   batch = (const int*)d_in[2];
    const float* w1  = (const float*)d_in[3];
    const float* b1  = (const float*)d_in[4];
    const float* w2  = (const float*)d_in[5];
    const float* b2  = (const float*)d_in[6];
    const float* w3  = (const float*)d_in[7];
    const float* b3  = (const float*)d_in[8];
    const float* g1  = (const float*)d_in[9];
    const float* be1 = (const float*)d_in[10];
    const float* g2  = (const float*)d_in[11];
    const float* be2 = (const float*)d_in[12];
    const float* fcw = (const float*)d_in[13];
    const float* fcb = (const float*)d_in[14];

    const int N = in_sizes[0] / DDIM;
    const int E = in_sizes[1] / 2;
    const int G = out_size / NCLS;
    const int* esrc = ei;
    const int* edst = ei + E;

    // workspace layout (floats), 256-float aligned
    float* ws    = (float*)d_ws;
    size_t off   = 0;
    float* dinv  = ws + off; off += ((size_t)N + 255) & ~255ull;
    float* bufA  = ws + off; off += (size_t)N * DDIM;      // activations / aggregate
    float* bufB  = ws + off; off += (size_t)N * DDIM;      // gemm output
    float* stats = ws + off; off += 256;                   // col sums / sumsq
    float* coef  = ws + off; off += 256;                   // bn scale / shift
    float* pooled= ws + off; off += (size_t)G * DDIM;
    float* cnt   = ws + off; off += (size_t)G;
    (void)ws_size; (void)n_in;

    const int n4 = N * (DDIM / 4);
    dim3 blk256(256), blk128(128);
    int gN      = (N + 255) / 256;
    int gE      = (E + 255) / 256;
    int gN4     = (n4 + 255) / 256;
    int gGemm   = ((N + 15) / 16 + 7) / 8;
    int gScat   = (E + 7) / 8;             // 8 waves per 256-thread block
    int gPool   = (N + 7) / 8;
    int gBN     = (N + BN_ROWS - 1) / BN_ROWS;

    // ---- degree / dinv ----
    fill_f32_kernel<<<gN, blk256, 0, stream>>>(dinv, 1.0f, N);   // self-loop
    deg_accum_kernel<<<gE, blk256, 0, stream>>>(edst, dinv, E);
    deg_rsqrt_kernel<<<gN, blk256, 0, stream>>>(dinv, N);

    // ---- conv1 + relu ----
    gemm128_wmma_kernel<<<gGemm, blk256, 0, stream>>>(x, w1, bufB, N);
    selfloop_init_kernel<<<gN4, blk256, 0, stream>>>(bufB, dinv, b1, bufA, N);
    scatter_kernel<<<gScat, blk256, 0, stream>>>(bufB, esrc, edst, dinv, bufA, E);
    relu_kernel<<<gN4, blk256, 0, stream>>>(bufA, n4);

    // ---- conv2 + relu ----
    gemm128_wmma_kernel<<<gGemm, blk256, 0, stream>>>(bufA, w2, bufB, N);
    selfloop_init_kernel<<<gN4, blk256, 0, stream>>>(bufB, dinv, b2, bufA, N);
    scatter_kernel<<<gScat, blk256, 0, stream>>>(bufB, esrc, edst, dinv, bufA, E);
    relu_kernel<<<gN4, blk256, 0, stream>>>(bufA, n4);

    // ---- bn1 + relu ----
    fill_f32_kernel<<<1, blk256, 0, stream>>>(stats, 0.f, 256);
    bn_stats_kernel<<<gBN, blk128, 0, stream>>>(bufA, stats, N);
    bn_coef_kernel<<<1, blk128, 0, stream>>>(stats, g1, be1, coef, N);
    bn_apply_relu_kernel<<<gN4, blk256, 0, stream>>>(bufA, coef, N);

    // ---- conv3 + relu ----
    gemm128_wmma_kernel<<<gGemm, blk256, 0, stream>>>(bufA, w3, bufB, N);
    selfloop_init_kernel<<<gN4, blk256, 0, stream>>>(bufB, dinv, b3, bufA, N);
    scatter_kernel<<<gScat, blk256, 0, stream>>>(bufB, esrc, edst, dinv, bufA, E);
    relu_kernel<<<gN4, blk256, 0, stream>>>(bufA, n4);

    // ---- bn2 + relu ----
    fill_f32_kernel<<<1, blk256, 0, stream>>>(stats, 0.f, 256);
    bn_stats_kernel<<<gBN, blk128, 0, stream>>>(bufA, stats, N);
    bn_coef_kernel<<<1, blk128, 0, stream>>>(stats, g2, be2, coef, N);
    bn_apply_relu_kernel<<<gN4, blk256, 0, stream>>>(bufA, coef, N);

    // ---- mean pool + head ----
    int gZp = ((G * DDIM) + 255) / 256;
    fill_f32_kernel<<<gZp, blk256, 0, stream>>>(pooled, 0.f, G * DDIM);
    fill_f32_kernel<<<(G + 255) / 256, blk256, 0, stream>>>(cnt, 0.f, G);
    pool_kernel<<<gPool, blk256, 0, stream>>>(bufA, batch, pooled, cnt, N);
    head_kernel<<<(G + 63) / 64, dim3(64), 0, stream>>>(pooled, cnt, fcw, fcb,
                                                        (float*)d_out, G);
}